// YOLOXDetectionLoss_14010183319939
// MI455X (gfx1250) — compile-verified
//
#include <hip/hip_runtime.h>
#include <hip/hip_bf16.h>

#define NCLS 10
#define ATOT 33600
#define BSZ  16
#define GN   60
#define TPK  10
#define L0   25600
#define L01  32000

typedef float v2f __attribute__((ext_vector_type(2)));
typedef float v8f __attribute__((ext_vector_type(8)));

__device__ __forceinline__ void anchor_xy(int a, float& xc, float& yc, float& sa) {
    int Wd, base; float s;
    if (a < L0)       { Wd = 160; base = 0;   s = 8.f;  }
    else if (a < L01) { Wd = 80;  base = L0;  s = 16.f; }
    else              { Wd = 40;  base = L01; s = 32.f; }
    int la = a - base;
    int y = la / Wd, x = la - y * Wd;
    xc = (x + 0.5f) * s; yc = (y + 0.5f) * s; sa = s;
}

__device__ __forceinline__ const float* level_base(const float* p0, const float* p1,
                                                   const float* p2, int b, int a,
                                                   long& off, long& chs) {
    const float* p; int Wd, base;
    if (a < L0)       { p = p0; Wd = 160; base = 0;   }
    else if (a < L01) { p = p1; Wd = 80;  base = L0;  }
    else              { p = p2; Wd = 40;  base = L01; }
    int la = a - base;
    int y = la / Wd, x = la - y * Wd;
    chs = (long)Wd * Wd;
    off = (long)b * 15 * chs + (long)y * Wd + x;
    return p;
}

// ---------------- 1) decode: bbox + Daug (log-prob features, K padded to 12) ---------
__global__ void k_decode(const float* __restrict__ p0, const float* __restrict__ p1,
                         const float* __restrict__ p2,
                         float* __restrict__ bbox, float* __restrict__ daug) {
    int i = blockIdx.x * 256 + threadIdx.x;
    if (i >= BSZ * ATOT) return;
    int b = i / ATOT, a = i - b * ATOT;
    long off, chs;
    const float* p = level_base(p0, p1, p2, b, a, off, chs);
    float s;
    { float xc, yc; anchor_xy(a, xc, yc, s); }
    int Wd = (a < L0) ? 160 : (a < L01 ? 80 : 40);
    int base = (a < L0) ? 0 : (a < L01 ? L0 : L01);
    int la = a - base; int gy = la / Wd, gx = la - gy * Wd;

    float v0 = p[off], v1 = p[off + chs], v2 = p[off + 2 * chs], v3 = p[off + 3 * chs];
    float obj = p[off + 4 * chs];
    float* bb = bbox + (long)i * 4;
    bb[0] = (v0 + gx) * s;
    bb[1] = (v1 + gy) * s;
    bb[2] = expf(v2) * s;
    bb[3] = expf(v3) * s;

    float so = 1.f / (1.f + expf(-obj));
    float S = 0.f;
    float* dg = daug + (long)i * 12;
    #pragma unroll
    for (int c = 0; c < NCLS; c++) {
        float vc = p[off + (5 + c) * chs];
        float sc = 1.f / (1.f + expf(-vc));
        float pp = sqrtf(sc * so);
        float lp = logf(pp + 1e-12f);
        float lq = logf(1.f - pp + 1e-12f);
        dg[c] = lp - lq;
        S += lq;
    }
    dg[10] = S;
    dg[11] = 0.f;
}

// ---------------- 2) Taug: one-hot rows padded to 64 x 12 ----------------------------
__global__ void k_taug(const float* __restrict__ labels, float* __restrict__ taug) {
    int i = blockIdx.x * 256 + threadIdx.x;  // B*64
    if (i >= BSZ * 64) return;
    int b = i / 64, g = i - b * 64;
    float* t = taug + (long)i * 12;
    #pragma unroll
    for (int c = 0; c < 12; c++) t[c] = 0.f;
    if (g < GN) {
        int cg = (int)labels[((long)b * GN + g) * 5];
        if (cg >= 0 && cg < NCLS) t[cg] = 1.f;
        t[10] = 1.f;  // coefficient of S_a
    }
}

// ---------------- 3) per-anchor foreground mask --------------------------------------
__global__ void k_fg(const float* __restrict__ labels, int* __restrict__ fg) {
    __shared__ float gt[GN * 5];
    __shared__ int gv[GN];
    int b = blockIdx.y;
    for (int j = threadIdx.x; j < GN * 5; j += 256) gt[j] = labels[(long)b * GN * 5 + j];
    __syncthreads();
    if (threadIdx.x < GN) {
        float s = gt[threadIdx.x*5] + gt[threadIdx.x*5+1] + gt[threadIdx.x*5+2] +
                  gt[threadIdx.x*5+3] + gt[threadIdx.x*5+4];
        gv[threadIdx.x] = (s > 0.f) ? 1 : 0;
    }
    __syncthreads();
    int a = blockIdx.x * 256 + threadIdx.x;
    if (a >= ATOT) return;
    float xc, yc, sa; anchor_xy(a, xc, yc, sa);
    float rad = 2.5f * sa;
    int any = 0;
    for (int g = 0; g < GN; g++) {
        if (!gv[g]) continue;
        float gx = gt[g*5+1], gy = gt[g*5+2], gw = gt[g*5+3], gh = gt[g*5+4];
        float l  = xc - (gx - gw*0.5f), r  = (gx + gw*0.5f) - xc;
        float tt = yc - (gy - gh*0.5f), bb = (gy + gh*0.5f) - yc;
        int inb = fminf(fminf(l, r), fminf(tt, bb)) > 0.f;
        float cl = xc - gx + rad, cr = gx + rad - xc;
        float ct = yc - gy + rad, cb = gy + rad - yc;
        int inc = fminf(fminf(cl, cr), fminf(ct, cb)) > 0.f;
        any |= (inb | inc);
    }
    fg[(long)b * ATOT + a] = any;
}

// ---------------- 4) cost + IoU matrices via WMMA (one batch) ------------------------
// grid (263, 4), block 256 = 8 waves; wave -> one 16(g) x 16(a) tile
__global__ void k_cost(int b, const float* __restrict__ labels,
                       const float* __restrict__ taug, const float* __restrict__ daug,
                       const float* __restrict__ bbox, const int* __restrict__ fg,
                       float* __restrict__ cost, float* __restrict__ ioum) {
    __shared__ float gt[16 * 5];
    __shared__ float gval[16];
    int gtile = blockIdx.y, g0 = gtile * 16;
    if (threadIdx.x < 80) {
        int g = g0 + threadIdx.x / 5, f = threadIdx.x % 5;
        gt[threadIdx.x] = (g < GN) ? labels[((long)b * GN + g) * 5 + f] : 0.f;
    }
    __syncthreads();
    if (threadIdx.x < 16) {
        float s = gt[threadIdx.x*5] + gt[threadIdx.x*5+1] + gt[threadIdx.x*5+2] +
                  gt[threadIdx.x*5+3] + gt[threadIdx.x*5+4];
        gval[threadIdx.x] = ((g0 + (int)threadIdx.x) < GN && s > 0.f) ? 1.f : 0.f;
    }
    __syncthreads();

    int wave = threadIdx.x >> 5, lane = threadIdx.x & 31;
    int ctile = blockIdx.x * 8 + wave;
    if (ctile >= ATOT / 16) return;   // wave-uniform: EXEC stays all-ones for WMMA

    int n = lane & 15, hi = lane >> 4;
    int aCol = ctile * 16 + n;
    const float* trow = taug + ((long)b * 64 + g0 + n) * 12;
    const float* drow = daug + ((long)b * ATOT + aCol) * 12;

    v2f a0, a1, a2, b0, b1, b2;
    int kb = 2 * hi;
    a0.x = trow[kb];     a0.y = trow[kb + 1];
    b0.x = drow[kb];     b0.y = drow[kb + 1];
    a1.x = trow[kb + 4]; a1.y = trow[kb + 5];
    b1.x = drow[kb + 4]; b1.y = drow[kb + 5];
    a2.x = trow[kb + 8]; a2.y = trow[kb + 9];
    b2.x = drow[kb + 8]; b2.y = drow[kb + 9];

    v8f c = {0.f, 0.f, 0.f, 0.f, 0.f, 0.f, 0.f, 0.f};
    // D = Taug_tile (16x12) * Daug_tile^T (12x16), three K=4 steps
    c = __builtin_amdgcn_wmma_f32_16x16x4_f32(false, a0, false, b0, (short)0, c, false, false);
    c = __builtin_amdgcn_wmma_f32_16x16x4_f32(false, a1, false, b1, (short)0, c, false, false);
    c = __builtin_amdgcn_wmma_f32_16x16x4_f32(false, a2, false, b2, (short)0, c, false, false);

    float xc, yc, sa; anchor_xy(aCol, xc, yc, sa);
    float rad = 2.5f * sa;
    const float* pb = bbox + ((long)b * ATOT + aCol) * 4;
    float px = pb[0], py = pb[1], pw = pb[2], ph = pb[3];
    float p1x = px - pw*0.5f, p1y = py - ph*0.5f, p2x = px + pw*0.5f, p2y = py + ph*0.5f;
    float parea = pw * ph;
    float fgA = fg[(long)b * ATOT + aCol] ? 1.f : 0.f;

    #pragma unroll
    for (int r = 0; r < 8; r++) {
        int gi = hi * 8 + r;        // row inside tile (C layout: M=r / M=8+r)
        int g = g0 + gi;
        if (g >= GN) continue;
        float gx = gt[gi*5+1], gy = gt[gi*5+2], gw = gt[gi*5+3], gh = gt[gi*5+4];
        float valid = gval[gi];
        float a1x = gx - gw*0.5f, a1y = gy - gh*0.5f, a2x = gx + gw*0.5f, a2y = gy + gh*0.5f;
        float iw = fmaxf(fminf(a2x, p2x) - fmaxf(a1x, p1x), 0.f);
        float ih = fmaxf(fminf(a2y, p2y) - fmaxf(a1y, p1y), 0.f);
        float inter = iw * ih;
        float iou = inter / (gw * gh + parea - inter + 1e-8f);

        float l  = xc - a1x, rr = a2x - xc, tt = yc - a1y, bb2 = a2y - yc;
        bool inb = (fminf(fminf(l, rr), fminf(tt, bb2)) > 0.f) && (valid > 0.f);
        float cl = xc - gx + rad, cr = gx + rad - xc;
        float ct = yc - gy + rad, cb = gy + rad - yc;
        bool inc = (fminf(fminf(cl, cr), fminf(ct, cb)) > 0.f) && (valid > 0.f);

        float cst = -c[r]                               // cls_cost from WMMA
                  - 3.f * logf(iou + 1e-8f)
                  + ((inb && inc) ? 0.f : 100000.f)
                  + ((fgA > 0.f && valid > 0.f) ? 0.f : 1000000.f);
        cost[(long)g * ATOT + aCol] = cst;
        ioum[(long)g * ATOT + aCol] = iou;
    }
}

// ---------------- 5) per-GT top-k (one batch): dyn_k + selected anchors --------------
__global__ void k_topk(int b, const float* __restrict__ labels,
                       const float* __restrict__ cost, const float* __restrict__ ioum,
                       const int* __restrict__ fg, int* __restrict__ sel_idx) {
    int g = blockIdx.x;
    __shared__ float sval;
    if (threadIdx.x == 0) {
        float s = 0.f;
        for (int f = 0; f < 5; f++) s += labels[((long)b * GN + g) * 5 + f];
        sval = (s > 0.f) ? 1.f : 0.f;
    }
    __syncthreads();
    bool valid = sval > 0.f;

    float tio[TPK], tco[TPK]; int tci[TPK];
    #pragma unroll
    for (int k = 0; k < TPK; k++) { tio[k] = -1.f; tco[k] = 3.4e38f; tci[k] = 0x7fffffff; }

    const float* crow = cost + (long)g * ATOT;
    const float* irow = ioum + (long)g * ATOT;
    const int*   frow = fg + (long)b * ATOT;
    for (int a = threadIdx.x; a < ATOT; a += 256) {
        float cv = crow[a];
        float ivf = (valid && frow[a]) ? irow[a] : 0.f;
        if (ivf > tio[TPK - 1]) {
            int k = TPK - 1;
            while (k > 0 && ivf > tio[k - 1]) { tio[k] = tio[k - 1]; k--; }
            tio[k] = ivf;
        }
        if (cv < tco[TPK - 1] || (cv == tco[TPK - 1] && a < tci[TPK - 1])) {
            int k = TPK - 1;
            while (k > 0 && (cv < tco[k - 1] || (cv == tco[k - 1] && a < tci[k - 1]))) {
                tco[k] = tco[k - 1]; tci[k] = tci[k - 1]; k--;
            }
            tco[k] = cv; tci[k] = a;
        }
    }

    __shared__ float sio[256 * TPK];
    __shared__ float sco[256 * TPK];
    __shared__ int   sci[256 * TPK];
    #pragma unroll
    for (int k = 0; k < TPK; k++) {
        sio[threadIdx.x * TPK + k] = tio[k];
        sco[threadIdx.x * TPK + k] = tco[k];
        sci[threadIdx.x * TPK + k] = tci[k];
    }
    __syncthreads();
    for (int off = 128; off > 0; off >>= 1) {
        if ((int)threadIdx.x < off) {
            float mio[TPK], mco[TPK]; int mci[TPK];
            float* Aio = &sio[threadIdx.x * TPK]; float* Bio = &sio[(threadIdx.x + off) * TPK];
            float* Aco = &sco[threadIdx.x * TPK]; float* Bco = &sco[(threadIdx.x + off) * TPK];
            int*   Aci = &sci[threadIdx.x * TPK]; int*   Bci = &sci[(threadIdx.x + off) * TPK];
            int i = 0, j = 0;
            #pragma unroll
            for (int k = 0; k < TPK; k++) mio[k] = (Aio[i] >= Bio[j]) ? Aio[i++] : Bio[j++];
            i = 0; j = 0;
            #pragma unroll
            for (int k = 0; k < TPK; k++) {
                bool ta = (Aco[i] < Bco[j]) || (Aco[i] == Bco[j] && Aci[i] <= Bci[j]);
                if (ta) { mco[k] = Aco[i]; mci[k] = Aci[i]; i++; }
                else    { mco[k] = Bco[j]; mci[k] = Bci[j]; j++; }
            }
            #pragma unroll
            for (int k = 0; k < TPK; k++) { Aio[k] = mio[k]; Aco[k] = mco[k]; Aci[k] = mci[k]; }
        }
        __syncthreads();
    }
    if (threadIdx.x == 0) {
        float ss = 0.f;
        for (int k = 0; k < TPK; k++) ss += sio[k];
        int dynk = (int)ss; if (dynk < 1) dynk = 1;
        for (int k = 0; k < TPK; k++) {
            bool sel = (k < dynk) && valid;
            sel_idx[g * TPK + k] = sel ? sci[k] : -1;
        }
    }
}

// ---------------- 6) conflict counting + resolution (one batch) ----------------------
__global__ void k_zero(int* __restrict__ nm, int* __restrict__ mg) {
    int a = blockIdx.x * 256 + threadIdx.x;
    if (a < ATOT) { nm[a] = 0; mg[a] = 0x7fffffff; }
}

__global__ void k_scatter(const int* __restrict__ sel_idx, int* __restrict__ nm,
                          int* __restrict__ mg) {
    int i = blockIdx.x * 256 + threadIdx.x;
    if (i >= GN * TPK) return;
    int a = sel_idx[i];
    if (a >= 0) {
        atomicAdd(&nm[a], 1);
        atomicMin(&mg[a], i / TPK);
    }
}

__global__ void k_finalize(int b, const float* __restrict__ labels,
                           const float* __restrict__ cost, const float* __restrict__ ioum,
                           const int* __restrict__ nm, const int* __restrict__ mgbuf,
                           float* __restrict__ fgf, float* __restrict__ miou,
                           int* __restrict__ mcls, float* __restrict__ rtgt) {
    int a = blockIdx.x * 256 + threadIdx.x;
    if (a >= ATOT) return;
    long o = (long)b * ATOT + a;
    int n = nm[a];
    if (n == 0) {
        fgf[o] = 0.f; miou[o] = 0.f; mcls[o] = 0;
        rtgt[o*4] = 0.f; rtgt[o*4+1] = 0.f; rtgt[o*4+2] = 0.f; rtgt[o*4+3] = 0.f;
        return;
    }
    int g;
    if (n == 1) g = mgbuf[a];
    else {  // argmin over g of cost column (first minimum, like jnp.argmin)
        float best = cost[a]; g = 0;
        for (int gg = 1; gg < GN; gg++) {
            float cv = cost[(long)gg * ATOT + a];
            if (cv < best) { best = cv; g = gg; }
        }
    }
    fgf[o] = 1.f;
    miou[o] = ioum[(long)g * ATOT + a];
    const float* L = &labels[((long)b * GN + g) * 5];
    mcls[o] = (int)L[0];
    rtgt[o*4]   = L[1]; rtgt[o*4+1] = L[2];
    rtgt[o*4+2] = L[3]; rtgt[o*4+3] = L[4];
}

// ---------------- 7) fused loss reduction --------------------------------------------
__global__ void k_init(float* __restrict__ accum) {
    if (threadIdx.x < 4) accum[threadIdx.x] = 0.f;
}

__global__ void k_loss(const float* __restrict__ p0, const float* __restrict__ p1,
                       const float* __restrict__ p2, const float* __restrict__ bbox,
                       const float* __restrict__ fgf, const float* __restrict__ miou,
                       const int* __restrict__ mcls, const float* __restrict__ rtgt,
                       float* __restrict__ accum) {
    int i = blockIdx.x * 256 + threadIdx.x;
    float s_iou = 0.f, s_obj = 0.f, s_cls = 0.f, s_fg = 0.f;
    if (i < BSZ * ATOT) {
        int b = i / ATOT, a = i - b * ATOT;
        long off, chs;
        const float* p = level_base(p0, p1, p2, b, a, off, chs);
        float obj = p[off + 4 * chs];
        float t = fgf[i];
        float bce = fmaxf(obj, 0.f) - obj * t + log1pf(expf(-fabsf(obj)));
        float pr = 1.f / (1.f + expf(-obj));
        float pt = t * pr + (1.f - t) * (1.f - pr);
        float wA = t * 0.25f + (1.f - t) * 0.75f;
        float om = 1.f - pt;
        s_obj = bce * wA * om * om;
        s_fg = t;
        if (t > 0.f) {
            const float* pb = &bbox[(long)i * 4];
            const float* tb = &rtgt[(long)i * 4];
            float p1x = pb[0] - pb[2]*0.5f, p1y = pb[1] - pb[3]*0.5f;
            float p2x = pb[0] + pb[2]*0.5f, p2y = pb[1] + pb[3]*0.5f;
            float t1x = tb[0] - tb[2]*0.5f, t1y = tb[1] - tb[3]*0.5f;
            float t2x = tb[0] + tb[2]*0.5f, t2y = tb[1] + tb[3]*0.5f;
            float iw = fmaxf(fminf(p2x, t2x) - fmaxf(p1x, t1x), 0.f);
            float ih = fmaxf(fminf(p2y, t2y) - fmaxf(p1y, t1y), 0.f);
            float inter = iw * ih;
            float iou = inter / (pb[2]*pb[3] + tb[2]*tb[3] - inter + 1e-8f);
            s_iou = 1.f - iou * iou;
            int mc = mcls[i]; float mi = miou[i];
            #pragma unroll
            for (int c = 0; c < NCLS; c++) {
                float x = p[off + (5 + c) * chs];
                float tc = (c == mc) ? mi : 0.f;
                s_cls += fmaxf(x, 0.f) - x * tc + log1pf(expf(-fabsf(x)));
            }
        }
    }
    __shared__ float r0[256], r1[256], r2[256], r3[256];
    int tid = threadIdx.x;
    r0[tid] = s_iou; r1[tid] = s_obj; r2[tid] = s_cls; r3[tid] = s_fg;
    __syncthreads();
    for (int o = 128; o > 0; o >>= 1) {
        if (tid < o) {
            r0[tid] += r0[tid + o]; r1[tid] += r1[tid + o];
            r2[tid] += r2[tid + o]; r3[tid] += r3[tid + o];
        }
        __syncthreads();
    }
    if (tid == 0) {
        atomicAdd(&accum[0], r0[0]); atomicAdd(&accum[1], r1[0]);
        atomicAdd(&accum[2], r2[0]); atomicAdd(&accum[3], r3[0]);
    }
}

__global__ void k_final(const float* __restrict__ accum, float* __restrict__ out) {
    float nf = fmaxf(accum[3], 1.f);
    float li = accum[0] / nf, lo = accum[1] / nf, lc = accum[2] / nf;
    out[0] = 5.f * li + lo;  // REG_W * loss_iou + loss_obj
    out[1] = li;
    out[2] = lo;
    out[3] = lc;
}

// ---------------- host launcher -------------------------------------------------------
extern "C" void kernel_launch(void* const* d_in, const int* in_sizes, int n_in,
                              void* d_out, int out_size, void* d_ws, size_t ws_size,
                              hipStream_t stream) {
    const float* p0 = (const float*)d_in[0];
    const float* p1 = (const float*)d_in[1];
    const float* p2 = (const float*)d_in[2];
    const float* labels = (const float*)d_in[3];
    float* out = (float*)d_out;

    size_t off = 0;
    auto carve = [&](size_t bytes) -> void* {
        void* p = (char*)d_ws + off;
        off = (off + bytes + 255) & ~(size_t)255;
        return p;
    };
    float* bbox = (float*)carve((size_t)BSZ * ATOT * 4 * 4);
    float* daug = (float*)carve((size_t)BSZ * ATOT * 12 * 4);
    float* taug = (float*)carve((size_t)BSZ * 64 * 12 * 4);
    int*   fg   = (int*)  carve((size_t)BSZ * ATOT * 4);
    float* cost = (float*)carve((size_t)GN * ATOT * 4);
    float* ioum = (float*)carve((size_t)GN * ATOT * 4);
    int*   sel  = (int*)  carve((size_t)GN * TPK * 4);
    int*   nm   = (int*)  carve((size_t)ATOT * 4);
    int*   mg   = (int*)  carve((size_t)ATOT * 4);
    float* fgf  = (float*)carve((size_t)BSZ * ATOT * 4);
    float* miou = (float*)carve((size_t)BSZ * ATOT * 4);
    int*   mcls = (int*)  carve((size_t)BSZ * ATOT * 4);
    float* rtgt = (float*)carve((size_t)BSZ * ATOT * 4 * 4);
    float* accum = (float*)carve(4 * 4);

    int nBA = BSZ * ATOT;
    int blkBA = (nBA + 255) / 256;
    int blkA  = (ATOT + 255) / 256;

    k_init<<<1, 64, 0, stream>>>(accum);
    k_decode<<<blkBA, 256, 0, stream>>>(p0, p1, p2, bbox, daug);
    k_taug<<<(BSZ * 64 + 255) / 256, 256, 0, stream>>>(labels, taug);
    k_fg<<<dim3(blkA, BSZ), 256, 0, stream>>>(labels, fg);

    int ctiles = ATOT / 16;                 // 2100
    int cblk = (ctiles + 7) / 8;            // 263 blocks of 8 waves
    for (int b = 0; b < BSZ; b++) {
        k_cost<<<dim3(cblk, 4), 256, 0, stream>>>(b, labels, taug, daug, bbox, fg, cost, ioum);
        k_topk<<<GN, 256, 0, stream>>>(b, labels, cost, ioum, fg, sel);
        k_zero<<<blkA, 256, 0, stream>>>(nm, mg);
        k_scatter<<<(GN * TPK + 255) / 256, 256, 0, stream>>>(sel, nm, mg);
        k_finalize<<<blkA, 256, 0, stream>>>(b, labels, cost, ioum, nm, mg,
                                             fgf, miou, mcls, rtgt);
    }
    k_loss<<<blkBA, 256, 0, stream>>>(p0, p1, p2, bbox, fgf, miou, mcls, rtgt, accum);
    k_final<<<1, 1, 0, stream>>>(accum, out);
}